// GCN_MPNN_35880156791371
// MI455X (gfx1250) — compile-verified
//
#include <hip/hip_runtime.h>

typedef __attribute__((ext_vector_type(2))) float v2f;
typedef __attribute__((ext_vector_type(8))) float v8f;

static constexpr int C = 128;   // IN_C == OUT_C == 128

// ---------------------------------------------------------------------------
// 1) degree init: deg[i] = 1.0 (self loop)
__global__ void k_deg_init(float* __restrict__ deg, int n) {
    int i = blockIdx.x * blockDim.x + threadIdx.x;
    if (i < n) deg[i] = 1.0f;
}

// 2) degree count over edge destinations (col)
__global__ void k_deg_count(const long long* __restrict__ col,
                            float* __restrict__ deg, int E) {
    int e = blockIdx.x * blockDim.x + threadIdx.x;
    if (e < E) atomicAdd(&deg[(int)col[e]], 1.0f);
}

// 3) in-place deg -> deg^{-1/2}
__global__ void k_rsqrt(float* __restrict__ deg, int n) {
    int i = blockIdx.x * blockDim.x + threadIdx.x;
    if (i < n) deg[i] = rsqrtf(deg[i]);
}

// ---------------------------------------------------------------------------
// 4) h = x @ W^T via V_WMMA_F32_16X16X4_F32 (full fp32 precision)
//    grid.x = M/16 tiles (M=10000 -> 625), block = 256 = 8 waves,
//    wave w computes N-tile [16w, 16w+16).
__global__ void __launch_bounds__(256)
k_gemm_wmma(const float* __restrict__ x, const float* __restrict__ W,
            float* __restrict__ h, int M) {
    const int lane = threadIdx.x & 31;
    const int wave = threadIdx.x >> 5;      // 0..7 -> N tile
    const int lo   = lane & 15;
    const int hi   = lane >> 4;             // 0 or 1

    const int m0 = blockIdx.x * 16;
    const int n0 = wave * 16;

    // clamp row for A loads on a (non-existent here) ragged tail; store is guarded
    const int mrow = (m0 + lo < M) ? (m0 + lo) : (M - 1);

    // A fragment source: x[m0+lo][k + 2*hi + {0,1}]
    const float* arow = x + (size_t)mrow * C + 2 * hi;
    // B fragment source: W[n0+lo][k + 2*hi + {0,1}]   (B = W^T)
    const float* brow = W + (size_t)(n0 + lo) * C + 2 * hi;

    v8f c = {};
#pragma unroll
    for (int k = 0; k < C; k += 4) {
        v2f a = *(const v2f*)(arow + k);
        v2f b = *(const v2f*)(brow + k);
        // 8 args: (neg_a, A, neg_b, B, c_mod, C, reuse_a, reuse_b)
        c = __builtin_amdgcn_wmma_f32_16x16x4_f32(
                false, a, false, b, (short)0, c, false, false);
    }

    // D layout: c[i] = D[i + 8*hi][lo]
#pragma unroll
    for (int i = 0; i < 8; ++i) {
        int m = m0 + i + 8 * hi;
        if (m < M) h[(size_t)m * C + n0 + lo] = c[i];
    }
}

// ---------------------------------------------------------------------------
// 5) out = dinv[n]^2 * h[n]  + bias      (self-loop message fused with bias)
__global__ void k_out_init(const float* __restrict__ h,
                           const float* __restrict__ dinv,
                           const float* __restrict__ bias,
                           float* __restrict__ out, int total) {
    int idx = blockIdx.x * blockDim.x + threadIdx.x;
    if (idx < total) {
        int node = idx >> 7;        // / 128
        int ch   = idx & (C - 1);
        float d  = dinv[node];
        out[idx] = d * d * h[idx] + bias[ch];
    }
}

// 6) edge scatter: one wave per edge, float4 per lane (covers 128 channels),
//    non-returning global_atomic_add_f32 into out[col].
__global__ void __launch_bounds__(256)
k_scatter(const long long* __restrict__ row, const long long* __restrict__ col,
          const float* __restrict__ h, const float* __restrict__ dinv,
          float* __restrict__ out, int E) {
    int e = blockIdx.x * 8 + (threadIdx.x >> 5);
    if (e >= E) return;
    int lane = threadIdx.x & 31;
    int r = (int)row[e];
    int c = (int)col[e];
    float w = dinv[r] * dinv[c];

    const float4 hv = *(const float4*)(h + (size_t)r * C + lane * 4);
    float* o = out + (size_t)c * C + lane * 4;
    atomicAdd(o + 0, w * hv.x);
    atomicAdd(o + 1, w * hv.y);
    atomicAdd(o + 2, w * hv.z);
    atomicAdd(o + 3, w * hv.w);
}

// ---------------------------------------------------------------------------
extern "C" void kernel_launch(void* const* d_in, const int* in_sizes, int n_in,
                              void* d_out, int out_size, void* d_ws, size_t ws_size,
                              hipStream_t stream) {
    const float*      x    = (const float*)d_in[0];        // [N, 128]
    const float*      W    = (const float*)d_in[1];        // [128, 128]
    const float*      bias = (const float*)d_in[2];        // [128]
    const long long*  ei   = (const long long*)d_in[3];    // [2, E] int64

    float* out = (float*)d_out;                            // [N, 128]

    const int N = in_sizes[0] / C;
    const int E = in_sizes[3] / 2;

    // workspace: h [N*128 floats] | deg/dinv [N floats]
    float* h   = (float*)d_ws;
    float* deg = (float*)((char*)d_ws + (size_t)N * C * sizeof(float));

    const long long* row = ei;        // ei[0][:]
    const long long* col = ei + E;    // ei[1][:]

    k_deg_init <<<(N + 255) / 256, 256, 0, stream>>>(deg, N);
    k_deg_count<<<(E + 255) / 256, 256, 0, stream>>>(col, deg, E);
    k_rsqrt    <<<(N + 255) / 256, 256, 0, stream>>>(deg, N);

    k_gemm_wmma<<<(N + 15) / 16, 256, 0, stream>>>(x, W, h, N);

    k_out_init <<<(N * C + 255) / 256, 256, 0, stream>>>(h, deg, bias, out, N * C);
    k_scatter  <<<(E + 7) / 8, 256, 0, stream>>>(row, col, h, deg, out, E);
}